// TireOperatorPINN_10814727651970
// MI455X (gfx1250) — compile-verified
//
#include <hip/hip_runtime.h>
#include <math.h>

typedef __attribute__((ext_vector_type(2))) float v2f;
typedef __attribute__((ext_vector_type(8))) float v8f;

#define MPTS 50
#define BLK 128

// ---- workspace layout (float offsets) ----
// A is stored PAIR-PACKED for WMMA B-operand b64 loads:
//   Apk[(k>>1)*128 + 2*n + (k&1)] = Ainv[k][n]   (64x64, zero padded)
#define WS_A     0        // 4096 floats, pair-packed K_ZZ_inv
#define WS_Z     4096     // [5][64] Z transposed, zero padded
#define WS_INV   4416     // inv_ls[5]
#define WS_PRIOR 4421     // prior_var
#define WS_W1    4424     // 8x16  (W1/sigma1)
#define WS_B1    4552     // 16
#define WS_W2    4568     // 16x32 (W2/sigma2)
#define WS_B2    5080     // 32
#define WS_W3    5112     // 32x2
#define WS_B3    5176     // 2
// total 5178 floats (~20.7 KB)

#define SQRT5 2.2360679774997896f
#define FIVE_THIRDS 1.6666666666666667f

__global__ __launch_bounds__(256) void prep_kernel(
    const float* __restrict__ W1, const float* __restrict__ u1, const float* __restrict__ b1,
    const float* __restrict__ W2, const float* __restrict__ u2, const float* __restrict__ b2,
    const float* __restrict__ W3, const float* __restrict__ b3,
    const float* __restrict__ log_ls, const float* __restrict__ log_var,
    const float* __restrict__ Z_raw, float* __restrict__ ws)
{
    __shared__ float Zl[MPTS][8];
    __shared__ float M1[MPTS][52];
    __shared__ float M2[MPTS][52];
    __shared__ float fcol[MPTS];
    __shared__ float sinv[5];
    __shared__ float spiv;
    __shared__ float ssig1, ssig2;

    const int t = threadIdx.x;
    const int NT = blockDim.x;

    if (t < 5) { float v = 1.0f / (expf(log_ls[t]) + 1e-8f); sinv[t] = v; ws[WS_INV + t] = v; }
    if (t == 5) { ws[WS_PRIOR] = expf(log_var[0]); }

    const float zscale[5] = {0.25f, 0.2f, 0.08f, 400.0f, 10.0f};
    const float zshift[5] = {0.0f, 0.0f, 0.0f, 800.0f, 12.0f};
    for (int idx = t; idx < MPTS * 5; idx += NT) {
        int j = idx / 5, d = idx % 5;
        Zl[j][d] = tanhf(Z_raw[idx]) * zscale[d] + zshift[d];
    }
    __syncthreads();

    // Z^T padded to [5][64]
    for (int idx = t; idx < 5 * 64; idx += NT) {
        int d = idx >> 6, j = idx & 63;
        ws[WS_Z + idx] = (j < MPTS) ? Zl[j][d] : 0.0f;
    }

    // Build K_ZZ + 1e-4 I  and identity
    float pv = expf(log_var[0]);
    for (int idx = t; idx < MPTS * MPTS; idx += NT) {
        int i = idx / MPTS, j = idx % MPTS;
        float d2 = 0.0f;
#pragma unroll
        for (int d = 0; d < 5; ++d) { float df = (Zl[i][d] - Zl[j][d]) * sinv[d]; d2 += df * df; }
        float dd = d2 + 1e-8f;
        float dist = sqrtf(dd);
        float sf = SQRT5 * dist;
        float k = pv * (1.0f + sf + FIVE_THIRDS * dd) * expf(-sf);
        M1[i][j] = k + ((i == j) ? 1e-4f : 0.0f);
        M2[i][j] = (i == j) ? 1.0f : 0.0f;
    }
    __syncthreads();

    // Gauss-Jordan inversion (SPD + ridge -> no pivoting needed)
    for (int k = 0; k < MPTS; ++k) {
        if (t == 0) spiv = 1.0f / M1[k][k];
        __syncthreads();
        float piv = spiv;
        for (int j = t; j < MPTS; j += NT) { M1[k][j] *= piv; M2[k][j] *= piv; }
        __syncthreads();
        if (t < MPTS) fcol[t] = (t == k) ? 0.0f : M1[t][k];
        __syncthreads();
        for (int idx = t; idx < MPTS * MPTS; idx += NT) {
            int i = idx / MPTS, j = idx % MPTS;
            if (i == k) continue;
            float f = fcol[i];
            M1[i][j] -= f * M1[k][j];
            M2[i][j] -= f * M2[k][j];
        }
        __syncthreads();
    }

    // A = K_ZZ_inv, zero padded to 64x64, PAIR-PACKED for b64 B-operand loads
    for (int idx = t; idx < 64 * 64; idx += NT) {
        int k = idx >> 6, n = idx & 63;
        float v = (k < MPTS && n < MPTS) ? M2[k][n] : 0.0f;
        ws[WS_A + (k >> 1) * 128 + 2 * n + (k & 1)] = v;
    }

    // spectral norms (single-step power iteration, as reference)
    if (t == 0) {
        float tv[8], v[16];
#pragma unroll
        for (int i = 0; i < 8; ++i) { float s = 0; for (int h = 0; h < 16; ++h) s += W1[i*16+h] * u1[h]; tv[i] = s; }
        float n2 = 0;
#pragma unroll
        for (int h = 0; h < 16; ++h) { float s = 0; for (int i = 0; i < 8; ++i) s += W1[i*16+h] * tv[i]; v[h] = s; n2 += s * s; }
        float inv = 1.0f / (sqrtf(n2) + 1e-8f);
        float s2 = 0;
#pragma unroll
        for (int i = 0; i < 8; ++i) { float s = 0; for (int h = 0; h < 16; ++h) s += W1[i*16+h] * v[h] * inv; s2 += s * s; }
        ssig1 = sqrtf(s2) + 1e-8f;
    }
    if (t == 1) {
        float tv[16], v[32];
#pragma unroll
        for (int h = 0; h < 16; ++h) { float s = 0; for (int o = 0; o < 32; ++o) s += W2[h*32+o] * u2[o]; tv[h] = s; }
        float n2 = 0;
#pragma unroll
        for (int o = 0; o < 32; ++o) { float s = 0; for (int h = 0; h < 16; ++h) s += W2[h*32+o] * tv[h]; v[o] = s; n2 += s * s; }
        float inv = 1.0f / (sqrtf(n2) + 1e-8f);
        float s2 = 0;
#pragma unroll
        for (int h = 0; h < 16; ++h) { float s = 0; for (int o = 0; o < 32; ++o) s += W2[h*32+o] * v[o] * inv; s2 += s * s; }
        ssig2 = sqrtf(s2) + 1e-8f;
    }
    __syncthreads();
    float is1 = 1.0f / ssig1, is2 = 1.0f / ssig2;
    for (int i = t; i < 128; i += NT) ws[WS_W1 + i] = W1[i] * is1;
    for (int i = t; i < 16;  i += NT) ws[WS_B1 + i] = b1[i];
    for (int i = t; i < 512; i += NT) ws[WS_W2 + i] = W2[i] * is2;
    for (int i = t; i < 32;  i += NT) ws[WS_B2 + i] = b2[i];
    for (int i = t; i < 64;  i += NT) ws[WS_W3 + i] = W3[i];
    for (int i = t; i < 2;   i += NT) ws[WS_B3 + i] = b3[i];
}

__global__ __launch_bounds__(BLK) void main_kernel(
    const float* __restrict__ st, const float* __restrict__ ws,
    float* __restrict__ out, int B)
{
    __shared__ float sA[64 * 64];      // 16 KB, pair-packed
    __shared__ float sK[BLK][68];      // 34 KB, stride 68 -> conflict-free b64 lane loads
    __shared__ float sZ[5 * 64];
    __shared__ float sW1[128], sb1[16], sW2[512], sb2[32], sW3[64], sb3[2];
    __shared__ float sinv[5];
    __shared__ float sprior;
    __shared__ float sRed[BLK];

    const int t = threadIdx.x;
    for (int i = t; i < 4096; i += BLK) sA[i] = ws[WS_A + i];
    for (int i = t; i < 320;  i += BLK) sZ[i] = ws[WS_Z + i];
    for (int i = t; i < 128;  i += BLK) sW1[i] = ws[WS_W1 + i];
    for (int i = t; i < 512;  i += BLK) sW2[i] = ws[WS_W2 + i];
    if (t < 16) sb1[t] = ws[WS_B1 + t];
    if (t < 32) sb2[t] = ws[WS_B2 + t];
    if (t < 64) sW3[t] = ws[WS_W3 + t];
    if (t < 2)  sb3[t] = ws[WS_B3 + t];
    if (t < 5)  sinv[t] = ws[WS_INV + t];
    if (t == 5) sprior = ws[WS_PRIOR];

    const int s = blockIdx.x * BLK + t;
    const bool valid = (s < B);
    float x[8];
    if (valid) {
        const float4* p = (const float4*)(st + 8 * (long)s);
        float4 a = p[0], b = p[1];
        x[0] = a.x; x[1] = a.y; x[2] = a.z; x[3] = a.w;
        x[4] = b.x; x[5] = b.y; x[6] = b.z; x[7] = b.w;
    } else {
#pragma unroll
        for (int i = 0; i < 8; ++i) x[i] = 0.0f;
    }
    __syncthreads();

    // ---- feats + MLP (VALU) ----
    float f[8];
    f[0] = sinf(x[0]); f[1] = sinf(2.0f * x[0]);
    f[2] = x[1]; f[3] = x[1] * x[1] * x[1];
    f[4] = x[2]; f[5] = x[3] * (1.0f / 1000.0f);
    f[6] = x[4] * (1.0f / 20.0f); f[7] = x[5];

    float h1[16];
#pragma unroll
    for (int h = 0; h < 16; ++h) {
        float a = sb1[h];
#pragma unroll
        for (int i = 0; i < 8; ++i) a += f[i] * sW1[i * 16 + h];
        h1[h] = tanhf(a);
    }
    float h2[32];
#pragma unroll
    for (int o = 0; o < 32; ++o) {
        float a = sb2[o];
#pragma unroll
        for (int h = 0; h < 16; ++h) a += h1[h] * sW2[h * 32 + o];
        h2[o] = tanhf(a);
    }
    float d0 = sb3[0], d1 = sb3[1];
#pragma unroll
    for (int o = 0; o < 32; ++o) { d0 += h2[o] * sW3[2 * o]; d1 += h2[o] * sW3[2 * o + 1]; }
    if (valid) ((float2*)out)[s] = make_float2(d0, d1);

    // ---- k_xZ (Matern-5/2), padded to 64 ----
    const float pv = sprior;
    for (int j = 0; j < 64; ++j) {
        float k = 0.0f;
        if (j < MPTS && valid) {
            float d2 = 0.0f;
#pragma unroll
            for (int d = 0; d < 5; ++d) { float df = (x[d] - sZ[d * 64 + j]) * sinv[d]; d2 += df * df; }
            float dd = d2 + 1e-8f;
            float dist = sqrtf(dd);
            float sf = SQRT5 * dist;
            k = pv * (1.0f + sf + FIVE_THIRDS * dd) * expf(-sf);
        }
        sK[t][j] = k;
    }
    __syncthreads();

    // ---- T = K_tile(16x64) x A(64x64) via V_WMMA_F32_16X16X4_F32 ----
    // k0 outer / 4 accumulator tiles: each K-fragment loaded once,
    // 4 independent WMMA chains, all LDS operand fetches are b64.
    const int wave = t >> 5, lane = t & 31;
    const int half = lane >> 4, l15 = lane & 15;

#pragma unroll
    for (int rt = 0; rt < 2; ++rt) {
        const int rbase = wave * 32 + rt * 16;
        v8f c0 = {0.f,0.f,0.f,0.f,0.f,0.f,0.f,0.f};
        v8f c1 = c0, c2 = c0, c3 = c0;
        const float* arow = &sK[rbase + l15][0];
#pragma unroll
        for (int k0 = 0; k0 < 64; k0 += 4) {
            // A-operand: lanes 0-15 hold K=k0,k0+1 ; lanes 16-31 K=k0+2,k0+3 (row M=l15)
            v2f a = *(const v2f*)(arow + k0 + 2 * half);
            // B-operand: pair-packed A, one b64 per lane per tile
            const float* bp = &sA[((k0 >> 1) + half) * 128 + 2 * l15];
            v2f b0 = *(const v2f*)(bp +  0);
            v2f b1 = *(const v2f*)(bp + 32);
            v2f b2 = *(const v2f*)(bp + 64);
            v2f b3 = *(const v2f*)(bp + 96);
            c0 = __builtin_amdgcn_wmma_f32_16x16x4_f32(false, a, false, b0, (short)0, c0, false, false);
            c1 = __builtin_amdgcn_wmma_f32_16x16x4_f32(false, a, false, b1, (short)0, c1, false, false);
            c2 = __builtin_amdgcn_wmma_f32_16x16x4_f32(false, a, false, b2, (short)0, c2, false, false);
            c3 = __builtin_amdgcn_wmma_f32_16x16x4_f32(false, a, false, b3, (short)0, c3, false, false);
        }
        // red partials: D layout VGPR v -> row v (lanes 0-15) / row v+8 (lanes 16-31), col n0+l15
        float acc[8] = {0.f,0.f,0.f,0.f,0.f,0.f,0.f,0.f};
#pragma unroll
        for (int v = 0; v < 8; ++v) {
            const float* krow = &sK[rbase + 8 * half + v][l15];
            acc[v] += c0[v] * krow[0];
            acc[v] += c1[v] * krow[16];
            acc[v] += c2[v] * krow[32];
            acc[v] += c3[v] * krow[48];
        }
        // reduce over the 16-lane N dimension within each half-wave
#pragma unroll
        for (int v = 0; v < 8; ++v) {
            float a0 = acc[v];
            a0 += __shfl_xor(a0, 1, 32);
            a0 += __shfl_xor(a0, 2, 32);
            a0 += __shfl_xor(a0, 4, 32);
            a0 += __shfl_xor(a0, 8, 32);
            if (l15 == v) sRed[rbase + 8 * half + v] = a0;
        }
    }
    __syncthreads();

    if (valid) {
        float red = sRed[t];
        float var = pv - red;
        var = (var < 1e-6f) ? 1e-6f : var;
        out[2 * (long)B + s] = sqrtf(var);
    }
}

extern "C" void kernel_launch(void* const* d_in, const int* in_sizes, int n_in,
                              void* d_out, int out_size, void* d_ws, size_t ws_size,
                              hipStream_t stream) {
    const float* st      = (const float*)d_in[0];
    const float* W1      = (const float*)d_in[1];
    const float* u1      = (const float*)d_in[2];
    const float* b1      = (const float*)d_in[3];
    const float* W2      = (const float*)d_in[4];
    const float* u2      = (const float*)d_in[5];
    const float* b2      = (const float*)d_in[6];
    const float* W3      = (const float*)d_in[7];
    const float* b3      = (const float*)d_in[8];
    const float* log_ls  = (const float*)d_in[9];
    const float* log_var = (const float*)d_in[10];
    const float* Z_raw   = (const float*)d_in[11];
    float* ws  = (float*)d_ws;
    float* out = (float*)d_out;

    const int B = in_sizes[0] / 8;

    prep_kernel<<<1, 256, 0, stream>>>(W1, u1, b1, W2, u2, b2, W3, b3,
                                       log_ls, log_var, Z_raw, ws);
    main_kernel<<<(B + BLK - 1) / BLK, BLK, 0, stream>>>(st, ws, out, B);
}